// GATEncoder_36421322670241
// MI455X (gfx1250) — compile-verified
//
#include <hip/hip_runtime.h>
#include <math.h>

// Problem constants (match reference)
#define NPG   169      // nodes per graph
#define NB    2        // graphs
#define NH    4        // heads
#define CC    256      // hidden dim
#define NNODE (NB*NPG) // 338 total nodes
#define KP    172      // alpha K padded to multiple of 4

typedef __attribute__((ext_vector_type(2))) float v2f;
typedef __attribute__((ext_vector_type(8))) float v8f;

// ---------------------------------------------------------------------------
// Batched/strided fp32 WMMA GEMM:  Y = act(X*W + bias) [+ residual]
//   X: [M,KLOOP] row stride ldx   W: [K?,Nc] row stride ldw
//   Y: [M,Nc]    row stride ldy   grid.z batches via strides sx/sw/sy
// One wave -> one 16x16 output tile, V_WMMA_F32_16X16X4_F32, K as template.
// Branch-free ragged edges: A-row / B-col clamped (garbage only lands in
// D rows/cols that are never stored); ragged K handled by zero-padded A
// (alpha) plus clamped B row index (0 * valid = 0). KCLAMP only for K=172.
// ---------------------------------------------------------------------------
template<int KLOOP, bool KCLAMP>
__global__ __launch_bounds__(256)
void wmma_gemm_f32(const float* __restrict__ X, const float* __restrict__ W,
                   const float* __restrict__ bias, const float* __restrict__ res,
                   float* __restrict__ Y,
                   int M, int Kw, int Nc,
                   int ldx, int ldw, int ldy,
                   long sx, long sw, long sy,
                   int act /*0=none,1=relu*/)
{
    const int wave = threadIdx.x >> 5;
    const int lane = threadIdx.x & 31;
    const int half = lane >> 4;     // 0: lanes 0-15, 1: lanes 16-31
    const int r    = lane & 15;

    const int m0 = blockIdx.y * 32 + (wave >> 2) * 16;
    const int n0 = blockIdx.x * 64 + (wave & 3) * 16;

    const float* Xb = X + (size_t)blockIdx.z * sx;
    const float* Wb = W + (size_t)blockIdx.z * sw;
    float*       Yb = Y + (size_t)blockIdx.z * sy;
    const float* Rb = res ? (res + (size_t)blockIdx.z * sy) : nullptr;

    // loop-invariant clamps: OOB A rows / B cols only affect unsaved D lanes
    const int rowA = min(m0 + r, M - 1);
    const int colB = min(n0 + r, Nc - 1);

    const float* pA = Xb + (size_t)rowA * ldx;   // A: lane=M, vgpr pair = K
    v8f acc0 = {}, acc1 = {};

#pragma unroll 4
    for (int k0 = 0; k0 < KLOOP; k0 += 4) {
        const int ke = k0 + 2 * half;            // v0 holds K=ke, v1 K=ke+1
        v2f a = *(const v2f*)(pA + ke);          // global_load_b64
        const int r0 = KCLAMP ? min(ke,     Kw - 1) : ke;
        const int r1 = KCLAMP ? min(ke + 1, Kw - 1) : ke + 1;
        v2f b;
        b.x = Wb[(size_t)r0 * ldw + colB];
        b.y = Wb[(size_t)r1 * ldw + colB];
        if ((k0 & 4) == 0)
            acc0 = __builtin_amdgcn_wmma_f32_16x16x4_f32(
                       false, a, false, b, (short)0, acc0, false, false);
        else
            acc1 = __builtin_amdgcn_wmma_f32_16x16x4_f32(
                       false, a, false, b, (short)0, acc1, false, false);
    }
    v8f acc = acc0 + acc1;

    if (n0 + r < Nc) {
        const float bv = bias ? bias[colB] : 0.f;
#pragma unroll
        for (int i = 0; i < 8; ++i) {
            const int row = m0 + i + 8 * half;   // C/D: vgpr i -> row i+8*half
            if (row < M) {
                float v = acc[i] + bv;
                if (act == 1) v = fmaxf(v, 0.f);
                if (Rb) v += Rb[(size_t)row * ldy + colB];
                Yb[(size_t)row * ldy + colB] = v;
            }
        }
    }
}

// ---------------------------------------------------------------------------
// GATv2 attention scores + softmax (per target node, per head, per graph).
// logits[i] = att_h . leaky_relu(xl[i,h,:] + xr[j,h,:]) , i != j
// Writes normalized alpha row, zero-padded to KP=172 columns, for the
// dense-graph aggregation GEMM. Nonlinearity inside reduction -> VALU.
// ---------------------------------------------------------------------------
__global__ __launch_bounds__(256)
void attn_alpha(const float* __restrict__ xl, const float* __restrict__ xr,
                const float* __restrict__ att, float* __restrict__ alphab)
{
    const int blk = blockIdx.x;                  // b*NH*NPG + h*NPG + j
    const int j = blk % NPG;
    const int h = (blk / NPG) % NH;
    const int b = blk / (NPG * NH);
    const int t = threadIdx.x;

    __shared__ float xrj[CC];
    __shared__ float attv[CC];
    __shared__ float lg[NPG];
    __shared__ float red[256];

    const size_t nodej = (size_t)(b * NPG + j);
    xrj[t]  = xr[nodej * (NH * CC) + h * CC + t];
    attv[t] = att[h * CC + t];
    __syncthreads();

    const int wave = t >> 5, lane = t & 31;
    for (int i = wave; i < NPG; i += 8) {
        float p = 0.f;
        const float* xli = xl + (size_t)(b * NPG + i) * (NH * CC) + h * CC;
        for (int c = lane; c < CC; c += 32) {
            float v = xli[c] + xrj[c];
            v = (v > 0.f) ? v : 0.2f * v;        // leaky_relu
            p += attv[c] * v;
        }
        for (int off = 16; off; off >>= 1) p += __shfl_down(p, off, 32);
        if (lane == 0) lg[i] = (i == j) ? -__builtin_inff() : p;
    }
    __syncthreads();

    // softmax over the 169 logits (self edge is -inf -> alpha 0)
    float m = (t < NPG) ? lg[t] : -__builtin_inff();
    red[t] = m; __syncthreads();
    for (int s = 128; s; s >>= 1) { if (t < s) red[t] = fmaxf(red[t], red[t + s]); __syncthreads(); }
    const float mx = red[0];
    __syncthreads();
    const float a = (t < NPG) ? __expf(lg[t] - mx) : 0.f;
    red[t] = a; __syncthreads();
    for (int s = 128; s; s >>= 1) { if (t < s) red[t] += red[t + s]; __syncthreads(); }
    const float inv = 1.f / (red[0] + 1e-16f);
    if (t < KP) alphab[(size_t)blk * KP + t] = (t < NPG) ? a * inv : 0.f;
}

// ---------------------------------------------------------------------------
// Head-mean + bias + LayerNorm over C=256 (one block per node).
// ---------------------------------------------------------------------------
__global__ __launch_bounds__(256)
void headmean_ln(const float* __restrict__ hagg, const float* __restrict__ bias,
                 const float* __restrict__ g, const float* __restrict__ bt,
                 float* __restrict__ out)
{
    const int n = blockIdx.x, t = threadIdx.x;
    const float* p = hagg + (size_t)n * (NH * CC);
    float v = 0.25f * (p[t] + p[CC + t] + p[2 * CC + t] + p[3 * CC + t]) + bias[t];

    __shared__ float red[256];
    red[t] = v; __syncthreads();
    for (int s = 128; s; s >>= 1) { if (t < s) red[t] += red[t + s]; __syncthreads(); }
    const float mu = red[0] * (1.f / CC);
    __syncthreads();
    const float d = v - mu;
    red[t] = d * d; __syncthreads();
    for (int s = 128; s; s >>= 1) { if (t < s) red[t] += red[t + s]; __syncthreads(); }
    const float var = red[0] * (1.f / CC);
    out[(size_t)n * CC + t] = d * rsqrtf(var + 1e-5f) * g[t] + bt[t];
}

// ---------------------------------------------------------------------------
// Global mean pool per graph -> [B, C]
// ---------------------------------------------------------------------------
__global__ __launch_bounds__(256)
void mean_pool(const float* __restrict__ x, float* __restrict__ out)
{
    const int b = blockIdx.x, t = threadIdx.x;
    float s = 0.f;
    for (int i = 0; i < NPG; ++i) s += x[(size_t)(b * NPG + i) * CC + t];
    out[b * CC + t] = s * (1.f / (float)NPG);
}

// ---------------------------------------------------------------------------
extern "C" void kernel_launch(void* const* d_in, const int* in_sizes, int n_in,
                              void* d_out, int out_size, void* d_ws, size_t ws_size,
                              hipStream_t stream)
{
    const float* x    = (const float*)d_in[0];
    const float* Wl   = (const float*)d_in[1];
    const float* bl   = (const float*)d_in[2];
    const float* Wr   = (const float*)d_in[3];
    const float* br   = (const float*)d_in[4];
    const float* att  = (const float*)d_in[5];
    const float* bias = (const float*)d_in[6];
    const float* ln_g = (const float*)d_in[7];
    const float* ln_b = (const float*)d_in[8];
    const float* W1   = (const float*)d_in[9];
    const float* b1   = (const float*)d_in[10];
    const float* W2   = (const float*)d_in[11];
    const float* b2   = (const float*)d_in[12];
    (void)in_sizes; (void)n_in; (void)out_size; (void)ws_size;

    // workspace carve-out (fp32)
    float* ws   = (float*)d_ws;
    float* xlb  = ws; ws += (size_t)NNODE * NH * CC;          // 338x1024
    float* xrb  = ws; ws += (size_t)NNODE * NH * CC;          // 338x1024
    float* hagg = ws; ws += (size_t)NNODE * NH * CC;          // 338x1024
    float* alph = ws; ws += (size_t)NB * NH * NPG * KP;       // 2x4x169x172
    float* hb   = ws; ws += (size_t)NNODE * CC;               // 338x256
    float* tb   = ws; ws += (size_t)NNODE * 2 * CC;           // 338x512
    float* xb0  = ws; ws += (size_t)NNODE * CC;
    float* xb1  = ws; ws += (size_t)NNODE * CC;

    const dim3 blk(256);
    const int MT  = (NNODE + 31) / 32;  // 11 row tiles for M=338
    const int MTg = (NPG + 31) / 32;    // 6 row tiles for M=169

    const float* xin = x;
    for (int l = 0; l < 3; ++l) {
        float* xout = (l & 1) ? xb1 : xb0;

        // xl = x @ Wl[l] + bl[l]   (338x256 @ 256x1024)
        wmma_gemm_f32<256, false><<<dim3(16, MT, 1), blk, 0, stream>>>(
            xin, Wl + (size_t)l * CC * NH * CC, bl + (size_t)l * NH * CC, nullptr, xlb,
            NNODE, CC, NH * CC, CC, NH * CC, NH * CC, 0, 0, 0, 0);
        // xr = x @ Wr[l] + br[l]
        wmma_gemm_f32<256, false><<<dim3(16, MT, 1), blk, 0, stream>>>(
            xin, Wr + (size_t)l * CC * NH * CC, br + (size_t)l * NH * CC, nullptr, xrb,
            NNODE, CC, NH * CC, CC, NH * CC, NH * CC, 0, 0, 0, 0);

        // GATv2 scores + softmax -> normalized, zero-padded alpha rows
        attn_alpha<<<dim3(NB * NH * NPG), blk, 0, stream>>>(
            xlb, xrb, att + (size_t)l * NH * CC, alph);

        // aggregation: agg[b,h] = alpha[b,h] (169x172z) @ xl_head (169x256), WMMA
        for (int b = 0; b < NB; ++b) {
            wmma_gemm_f32<KP, true><<<dim3(4, MTg, NH), blk, 0, stream>>>(
                alph + (size_t)b * NH * NPG * KP,
                xlb + (size_t)b * NPG * NH * CC, nullptr, nullptr,
                hagg + (size_t)b * NPG * NH * CC,
                NPG, NPG, CC, KP, NH * CC, NH * CC,
                (long)NPG * KP, (long)CC, (long)CC, 0);
        }

        // head mean + bias + layernorm
        headmean_ln<<<dim3(NNODE), blk, 0, stream>>>(
            hagg, bias + (size_t)l * CC, ln_g + (size_t)l * CC, ln_b + (size_t)l * CC, hb);

        // MLP: t = relu(h @ W1 + b1)   (338x256 @ 256x512)
        wmma_gemm_f32<256, false><<<dim3(8, MT, 1), blk, 0, stream>>>(
            hb, W1 + (size_t)l * CC * 2 * CC, b1 + (size_t)l * 2 * CC, nullptr, tb,
            NNODE, CC, 2 * CC, CC, 2 * CC, 2 * CC, 0, 0, 0, 1);
        // x = residual + t @ W2 + b2   (338x512 @ 512x256)
        wmma_gemm_f32<512, false><<<dim3(4, MT, 1), blk, 0, stream>>>(
            tb, W2 + (size_t)l * 2 * CC * CC, b2 + (size_t)l * CC, xin, xout,
            NNODE, 2 * CC, CC, 2 * CC, CC, CC, 0, 0, 0, 0);

        xin = xout;
    }

    mean_pool<<<dim3(NB), blk, 0, stream>>>(xin, (float*)d_out);
}